// NeuralFeatMap_64278480552060
// MI455X (gfx1250) — compile-verified
//
#include <hip/hip_runtime.h>

typedef __attribute__((ext_vector_type(2))) float v2f;
typedef __attribute__((ext_vector_type(8))) float v8f;

#define N_PTS 150000
#define KDIM  64
#define CDIM  128
#define IMG_W 512
#define IMG_HW (512 * 512)

// ---------------------------------------------------------------------------
// Kernel 1: zero the [C, H, W] canvas with wide (b128) stores.
// ---------------------------------------------------------------------------
__global__ __launch_bounds__(256) void NeuralFeatMap_zero_kernel(float4* __restrict__ out, int n4) {
    int i = blockIdx.x * blockDim.x + threadIdx.x;
    if (i < n4) {
        out[i] = make_float4(0.f, 0.f, 0.f, 0.f);
    }
}

// ---------------------------------------------------------------------------
// Kernel 2: fp32 WMMA GEMM (feature[128,64] @ vertMask[64,N]) fused with the
// scatter into out[C, H, W].  One wave32 per 16-wide N tile; 8 C-tiles of 16;
// K=64 reduced with 16 x v_wmma_f32_16x16x4_f32.
// ---------------------------------------------------------------------------
__global__ __launch_bounds__(256) void NeuralFeatMap_gemm_scatter_kernel(
    const float* __restrict__ feature,   // [C, K] row-major
    const float* __restrict__ vertMask,  // [K, N] row-major
    const int*   __restrict__ pX,        // [N]
    const int*   __restrict__ pY,        // [N]
    float*       __restrict__ out)       // [C, H, W]
{
    const int lane = threadIdx.x & 31;
    const int wave = threadIdx.x >> 5;
    const int tile = blockIdx.x * 8 + wave;      // n-tile index (16 columns each)
    const int n0 = tile * 16;
    if (n0 >= N_PTS) return;                     // wave-uniform: EXEC stays all-1s

    const int half = lane >> 4;                  // 0: lanes 0-15, 1: lanes 16-31
    const int nl   = lane & 15;                  // column within tile / row within A

    // --- Preload all 16 K-step B fragments (vertMask[:, n0:n0+16]) ----------
    // B layout (32-bit, 4x16): VGPR0 = row k0+2*half, VGPR1 = row k0+2*half+1,
    // N striped across lanes -> coalesced 64B row segments per half-wave.
    v2f b[16];
#pragma unroll
    for (int ks = 0; ks < 16; ++ks) {
        const int krow = 4 * ks + 2 * half;
        b[ks].x = vertMask[(size_t)krow       * N_PTS + n0 + nl];
        b[ks].y = vertMask[(size_t)(krow + 1) * N_PTS + n0 + nl];
    }

    // --- Scatter target pixel for this lane's column ------------------------
    const int n   = n0 + nl;
    const int pix = pY[n] * IMG_W + pX[n];

    // --- 8 C-tiles of 16 rows each ------------------------------------------
#pragma unroll
    for (int mt = 0; mt < 8; ++mt) {
        const int m0 = mt * 16;
        v8f acc = {};                            // C = 0
#pragma unroll
        for (int ks = 0; ks < 16; ++ks) {
            const int krow = 4 * ks + 2 * half;
            // A layout (32-bit, 16x4): lane 16*half+m holds feature[m0+m, krow]
            // and feature[m0+m, krow+1] -> contiguous pair, single b64 load.
            const v2f a = *(const v2f*)(feature + (size_t)(m0 + nl) * KDIM + krow);
            acc = __builtin_amdgcn_wmma_f32_16x16x4_f32(
                /*neg_a=*/false, a, /*neg_b=*/false, b[ks],
                /*c_mod=*/(short)0, acc, /*reuse_a=*/false, /*reuse_b=*/false);
        }
        // D layout: lane 16*half+nl holds column nl, rows (v + 8*half).
        float* outp = out + ((size_t)(m0 + 8 * half)) * IMG_HW + pix;
#pragma unroll
        for (int v = 0; v < 8; ++v) {
            outp[(size_t)v * IMG_HW] = acc[v];
        }
    }
}

// ---------------------------------------------------------------------------
// Launch: inputs are (H, W, pX, pY, vertMask, feature) per setup_inputs order.
// ---------------------------------------------------------------------------
extern "C" void kernel_launch(void* const* d_in, const int* in_sizes, int n_in,
                              void* d_out, int out_size, void* d_ws, size_t ws_size,
                              hipStream_t stream) {
    const int*   pX       = (const int*)d_in[2];
    const int*   pY       = (const int*)d_in[3];
    const float* vertMask = (const float*)d_in[4];
    const float* feature  = (const float*)d_in[5];
    float*       out      = (float*)d_out;

    // Zero the 128 MB canvas (out_size = 128*512*512, divisible by 4).
    const int n4 = out_size / 4;
    NeuralFeatMap_zero_kernel<<<(n4 + 255) / 256, 256, 0, stream>>>((float4*)out, n4);

    // 9375 n-tiles of 16 columns; 8 waves (n-tiles) per 256-thread block.
    const int ntiles = N_PTS / 16;               // 9375 (exact)
    const int blocks = (ntiles + 7) / 8;         // 1172
    NeuralFeatMap_gemm_scatter_kernel<<<blocks, 256, 0, stream>>>(
        feature, vertMask, pX, pY, out);
}